// SGModule_1949915152401
// MI455X (gfx1250) — compile-verified
//
#include <hip/hip_runtime.h>
#include <hip/hip_bf16.h>

// ---------------------------------------------------------------------------
// SGFormer module for MI455X (gfx1250).  fp32 end-to-end; all GEMMs use
// V_WMMA_F32_16X16X4_F32 (the CDNA5 fp32 matrix op).  wave32 everywhere.
// B tiles are staged LDS-direct with global_load_async_to_lds_b128 (ASYNCcnt)
// and double-buffered; A tiles are software-pipelined through VGPRs.
// ---------------------------------------------------------------------------

#define G_NUM   32
#define MAXN    2304
#define CIN     256
#define CH      512
#define ROWS    (G_NUM * MAXN)      // 73728
#define NLAYER  2

typedef __attribute__((ext_vector_type(2))) float v2f;
typedef __attribute__((ext_vector_type(8))) float v8f;
typedef __attribute__((ext_vector_type(4))) int   v4i;

#if __has_builtin(__builtin_amdgcn_global_load_async_to_lds_b128)
#define HAVE_ASYNC_LDS 1
typedef __attribute__((address_space(1))) v4i gas_v4i;   // global
typedef __attribute__((address_space(3))) v4i las_v4i;   // LDS
#else
#define HAVE_ASYNC_LDS 0
#endif

__device__ __forceinline__ void async_copy16(const float* gsrc, float* ldst)
{
#if HAVE_ASYNC_LDS
    __builtin_amdgcn_global_load_async_to_lds_b128(
        (gas_v4i*)gsrc, (las_v4i*)ldst, 0, 0);
#else
    *(float4*)ldst = *(const float4*)gsrc;
#endif
}

__device__ __forceinline__ void async_wait0()
{
#if HAVE_ASYNC_LDS
#if __has_builtin(__builtin_amdgcn_s_wait_asynccnt)
    __builtin_amdgcn_s_wait_asynccnt(0);
#else
    asm volatile("s_wait_asynccnt 0x0" ::: "memory");
#endif
#endif
}

// ---------------------------------------------------------------------------
// Kernel 0: deterministic per-graph stable positions.
// node_of[g*MAXN + p] = original row index (or -1 for padding).
// ---------------------------------------------------------------------------
__global__ __launch_bounds__(256)
void build_index_kernel(const int* __restrict__ batch, int N,
                        int* __restrict__ node_of, int* __restrict__ counts)
{
    const int g    = blockIdx.x;
    const int tid  = threadIdx.x;
    const int lane = tid & 31;
    const int w    = tid >> 5;
    __shared__ int warp_tot[8];

    int base = 0;
    for (int start = 0; start < N; start += 256) {
        const int i   = start + tid;
        const int flg = (i < N && batch[i] == g) ? 1 : 0;
        unsigned long long mb = __ballot(flg);
        unsigned mw = (unsigned)mb;                    // wave32 mask
        int pre = __popc(mw & ((1u << lane) - 1u));
        int tot = __popc(mw);
        if (lane == 0) warp_tot[w] = tot;
        __syncthreads();
        int woff = 0, btot = 0;
        #pragma unroll
        for (int j = 0; j < 8; ++j) {
            int s = warp_tot[j];
            if (j < w) woff += s;
            btot += s;
        }
        if (flg) node_of[g * MAXN + base + woff + pre] = i;
        base += btot;
        __syncthreads();
    }
    if (tid == 0) counts[g] = base;
    for (int p = base + tid; p < MAXN; p += 256)
        node_of[g * MAXN + p] = -1;
}

// ---------------------------------------------------------------------------
// Templated WMMA GEMM:  C[M x N] = op(A)[M x Kd] @ B[Kd x N]
//   MODE 0: op(A)[m][k] = A[m*lda + k]               (plain, optional batch)
//   MODE 1: r = node_of[m]; op(A)[m][k] = r>=0 ? A[r*lda + k] : 0   (fc gather)
//   MODE 2: op(A)[m][k] = A[k*lda + m]               (K^T V, batched)
// Block: 256 threads = 8 waves.  Tile 64(M) x 256(N), K-chunks of 16,
// double-buffered.  Each wave owns a 32x64 sub-tile: 2 bands x 4 col-tiles =
// 8 x v_wmma_f32_16x16x4_f32 accumulators; B fragments reused across bands.
// ---------------------------------------------------------------------------
template <int MODE>
__global__ __launch_bounds__(256)
void gemm_wmma_f32(const float* __restrict__ A, const float* __restrict__ B,
                   float* __restrict__ C, int M, int N, int Kd, int lda,
                   long strideAz, long strideBz, long strideCz,
                   const int* __restrict__ node_of)
{
    __shared__ float As[2][64 * 18];  // padded stride 18 -> conflict-free reads
    __shared__ float Bs[2][16 * 256];

    const float* Ab = A + (long)blockIdx.z * strideAz;
    const float* Bb = B + (long)blockIdx.z * strideBz;
    float*       Cb = C + (long)blockIdx.z * strideCz;

    const int am0  = blockIdx.x * 64;
    const int nb0  = blockIdx.y * 256;
    const int tid  = threadIdx.x;
    const int lane = tid & 31;
    const int wave = tid >> 5;
    const int wseg = wave & 1;        // which 32-row segment of the 64-row tile
    const int ngrp = wave >> 1;       // which 64-col group of the 256-col tile
    const int half = lane >> 4;       // lanes 16..31 hold K+2 / M+8 per ISA layout
    const int l16  = lane & 15;

    // ---- hoist per-thread A source pointers / LDS slots out of the K loop ----
    const float* aptr[4];
    int          asl[4];
    long         astep;               // per-chunk advance (elements)
    if (MODE == 2) {
        const int m = tid & 63;
        #pragma unroll
        for (int it = 0; it < 4; ++it) {
            const int k = (tid >> 6) + it * 4;
            aptr[it] = Ab + (long)k * lda + (am0 + m);
            asl[it]  = m * 18 + k;
        }
        astep = (long)16 * lda;
    } else {
        const int k = tid & 15;
        #pragma unroll
        for (int it = 0; it < 4; ++it) {
            const int m = (tid >> 4) + it * 16;
            asl[it] = m * 18 + k;
            if (MODE == 1) {
                const int r = node_of[am0 + m];
                aptr[it] = (r >= 0) ? (Ab + (long)r * lda + k) : nullptr;
            } else {
                aptr[it] = Ab + (long)(am0 + m) * lda + k;
            }
        }
        astep = 16;
    }

    v8f acc[2][4];
    #pragma unroll
    for (int mb = 0; mb < 2; ++mb)
        #pragma unroll
        for (int nt = 0; nt < 4; ++nt)
            acc[mb][nt] = (v8f){};

    const int nch = Kd >> 4;

    // ---- prologue: stage chunk 0 into buffer 0 ----
    {
        #pragma unroll
        for (int it = 0; it < 4; ++it) {
            float v;
            if (MODE == 1) v = aptr[it] ? *aptr[it] : 0.0f;
            else           v = *aptr[it];
            As[0][asl[it]] = v;
        }
        #pragma unroll
        for (int it = 0; it < 4; ++it) {
            const int u = tid + it * 256;
            const int r = u >> 6, c4 = u & 63;
            async_copy16(Bb + (long)r * N + nb0 + (c4 << 2),
                         &Bs[0][r * 256 + (c4 << 2)]);
        }
    }
    async_wait0();
    __syncthreads();

    for (int i = 0; i < nch; ++i) {
        const int p = i & 1;

        // ---- issue next chunk: A global loads first (latency hides under
        //      the WMMAs below), B via async LDS-direct copies ----
        float areg[4];
        const bool pre = (i + 1 < nch);
        if (pre) {
            const long aoff = (long)(i + 1) * astep;
            #pragma unroll
            for (int it = 0; it < 4; ++it) {
                if (MODE == 1) areg[it] = aptr[it] ? aptr[it][aoff] : 0.0f;
                else           areg[it] = aptr[it][aoff];
            }
            const long k0 = (long)(i + 1) * 16;
            #pragma unroll
            for (int it = 0; it < 4; ++it) {
                const int u = tid + it * 256;
                const int r = u >> 6, c4 = u & 63;
                async_copy16(Bb + (k0 + r) * N + nb0 + (c4 << 2),
                             &Bs[p ^ 1][r * 256 + (c4 << 2)]);
            }
        }

        // ---- compute on buffer p ----
        #pragma unroll
        for (int ks = 0; ks < 4; ++ks) {
            const int kk = ks * 4 + half * 2;     // fragment K offset
            v2f a0, a1;
            a0.x = As[p][(wseg * 32 + l16) * 18 + kk];
            a0.y = As[p][(wseg * 32 + l16) * 18 + kk + 1];
            a1.x = As[p][(wseg * 32 + 16 + l16) * 18 + kk];
            a1.y = As[p][(wseg * 32 + 16 + l16) * 18 + kk + 1];
            #pragma unroll
            for (int nt = 0; nt < 4; ++nt) {
                const int col = ngrp * 64 + nt * 16 + l16;
                v2f b;
                b.x = Bs[p][kk * 256 + col];
                b.y = Bs[p][(kk + 1) * 256 + col];
                acc[0][nt] = __builtin_amdgcn_wmma_f32_16x16x4_f32(
                                 false, a0, false, b, (short)0, acc[0][nt],
                                 false, false);
                acc[1][nt] = __builtin_amdgcn_wmma_f32_16x16x4_f32(
                                 false, a1, false, b, (short)0, acc[1][nt],
                                 false, false);
            }
        }

        // ---- park next A tile in LDS, then fence the async B copies ----
        if (pre) {
            #pragma unroll
            for (int it = 0; it < 4; ++it)
                As[p ^ 1][asl[it]] = areg[it];
        }
        async_wait0();
        __syncthreads();
    }

    // ---- store D: VGPR v -> row v (lanes 0-15) / v+8 (lanes 16-31) ----
    #pragma unroll
    for (int mb = 0; mb < 2; ++mb) {
        #pragma unroll
        for (int nt = 0; nt < 4; ++nt) {
            const int col = nb0 + ngrp * 64 + nt * 16 + l16;
            #pragma unroll
            for (int v = 0; v < 8; ++v) {
                const int row = am0 + wseg * 32 + mb * 16 + half * 8 + v;
                Cb[(long)row * N + col] = acc[mb][nt][v];
            }
        }
    }
}

// ---------------------------------------------------------------------------
// LayerNorm(+bias) + ReLU over 512-wide rows.  One block per row.
// ---------------------------------------------------------------------------
__global__ __launch_bounds__(256)
void ln_relu_kernel(const float* __restrict__ in, const float* __restrict__ bias,
                    const float* __restrict__ gw, const float* __restrict__ bw,
                    float* __restrict__ out, int hasBias)
{
    __shared__ float red[256];
    const long row = blockIdx.x;
    const int tid  = threadIdx.x;
    const long off = row * CH;

    float v0 = in[off + tid]       + (hasBias ? bias[tid]       : 0.0f);
    float v1 = in[off + tid + 256] + (hasBias ? bias[tid + 256] : 0.0f);

    red[tid] = v0 + v1;
    __syncthreads();
    for (int s = 128; s > 0; s >>= 1) {
        if (tid < s) red[tid] += red[tid + s];
        __syncthreads();
    }
    const float mean = red[0] * (1.0f / CH);
    __syncthreads();

    const float d0 = v0 - mean, d1 = v1 - mean;
    red[tid] = d0 * d0 + d1 * d1;
    __syncthreads();
    for (int s = 128; s > 0; s >>= 1) {
        if (tid < s) red[tid] += red[tid + s];
        __syncthreads();
    }
    const float var = red[0] * (1.0f / CH);
    const float rs  = rsqrtf(var + 1e-5f);

    out[off + tid]       = fmaxf(d0 * rs * gw[tid]       + bw[tid],       0.0f);
    out[off + tid + 256] = fmaxf(d1 * rs * gw[tid + 256] + bw[tid + 256], 0.0f);
}

// ---------------------------------------------------------------------------
// Zero V rows of padding slots (vs = where(mask, vs, 0)).
// ---------------------------------------------------------------------------
__global__ __launch_bounds__(128)
void maskv_kernel(float* __restrict__ V, const int* __restrict__ node_of)
{
    const int row = blockIdx.x;
    if (node_of[row] >= 0) return;
    float4 z = make_float4(0.f, 0.f, 0.f, 0.f);
    *(float4*)(V + (long)row * CH + threadIdx.x * 4) = z;
}

// ---------------------------------------------------------------------------
// Two-stage deterministic sum-of-squares over Q and K (global Frobenius norms).
// ---------------------------------------------------------------------------
__global__ __launch_bounds__(256)
void sumsq2_kernel(const float* __restrict__ Q, const float* __restrict__ K,
                   float* __restrict__ part)
{
    __shared__ float red[256];
    const long total4 = (long)ROWS * CH / 4;      // 9437184
    const long per    = total4 / 1024;            // 9216
    const long base   = (long)blockIdx.x * per;
    const int  tid    = threadIdx.x;

    float sq = 0.0f, sk = 0.0f;
    for (long i = base + tid; i < base + per; i += 256) {
        float4 a = ((const float4*)Q)[i];
        float4 b = ((const float4*)K)[i];
        sq += a.x * a.x + a.y * a.y + a.z * a.z + a.w * a.w;
        sk += b.x * b.x + b.y * b.y + b.z * b.z + b.w * b.w;
    }
    red[tid] = sq;
    __syncthreads();
    for (int s = 128; s > 0; s >>= 1) {
        if (tid < s) red[tid] += red[tid + s];
        __syncthreads();
    }
    if (tid == 0) part[blockIdx.x] = red[0];
    __syncthreads();
    red[tid] = sk;
    __syncthreads();
    for (int s = 128; s > 0; s >>= 1) {
        if (tid < s) red[tid] += red[tid + s];
        __syncthreads();
    }
    if (tid == 0) part[1024 + blockIdx.x] = red[0];
}

__global__ __launch_bounds__(256)
void finalize_scale_kernel(const float* __restrict__ part, float* __restrict__ scal)
{
    __shared__ float red[256];
    const int tid = threadIdx.x;
    float sq = 0.0f, sk = 0.0f;
    for (int i = tid; i < 1024; i += 256) {
        sq += part[i];
        sk += part[1024 + i];
    }
    red[tid] = sq;
    __syncthreads();
    for (int s = 128; s > 0; s >>= 1) {
        if (tid < s) red[tid] += red[tid + s];
        __syncthreads();
    }
    const float sqt = red[0];
    __syncthreads();
    red[tid] = sk;
    __syncthreads();
    for (int s = 128; s > 0; s >>= 1) {
        if (tid < s) red[tid] += red[tid + s];
        __syncthreads();
    }
    if (tid == 0) scal[0] = 1.0f / (sqrtf(sqt) * sqrtf(red[0]));
}

// ---------------------------------------------------------------------------
// ks_sum[g][c] = sum_p K[g,p,c]  (unmasked, per reference).
// ---------------------------------------------------------------------------
__global__ __launch_bounds__(256)
void ksum_kernel(const float* __restrict__ K, float* __restrict__ ksum)
{
    const int g = blockIdx.x;
    const int c = blockIdx.y * 256 + threadIdx.x;
    const float* p = K + (long)g * MAXN * CH + c;
    float s = 0.0f;
    for (int i = 0; i < MAXN; ++i) s += p[(long)i * CH];
    ksum[g * CH + c] = s;
}

// ---------------------------------------------------------------------------
// Attention combine + residual + LayerNorm + ReLU (one block per row):
//   denom = dot(Q[row], ksum[g]) * scl + n
//   a     = (S[row] * scl + n * V[row]) / denom
//   Hn    = relu(LN((a + H[row]) / 2; g, b))
// ---------------------------------------------------------------------------
__global__ __launch_bounds__(256)
void attn_ln_kernel(const float* __restrict__ S, const float* __restrict__ Q,
                    const float* __restrict__ V, const float* __restrict__ H,
                    const float* __restrict__ ksum, const float* __restrict__ scal,
                    const float* __restrict__ gw, const float* __restrict__ bw,
                    float* __restrict__ Hn)
{
    __shared__ float red[256];
    const long row = blockIdx.x;
    const int  g   = (int)(row / MAXN);
    const int  tid = threadIdx.x;
    const long off = row * CH;
    const float scl = scal[0];
    const float nn  = (float)MAXN;

    const float q0 = Q[off + tid], q1 = Q[off + tid + 256];
    red[tid] = q0 * ksum[g * CH + tid] + q1 * ksum[g * CH + tid + 256];
    __syncthreads();
    for (int s = 128; s > 0; s >>= 1) {
        if (tid < s) red[tid] += red[tid + s];
        __syncthreads();
    }
    const float denom = red[0] * scl + nn;
    __syncthreads();

    const float a0 = (S[off + tid]       * scl + nn * V[off + tid])       / denom;
    const float a1 = (S[off + tid + 256] * scl + nn * V[off + tid + 256]) / denom;
    const float t0 = (a0 + H[off + tid])       * 0.5f;
    const float t1 = (a1 + H[off + tid + 256]) * 0.5f;

    red[tid] = t0 + t1;
    __syncthreads();
    for (int s = 128; s > 0; s >>= 1) {
        if (tid < s) red[tid] += red[tid + s];
        __syncthreads();
    }
    const float mean = red[0] * (1.0f / CH);
    __syncthreads();

    const float d0 = t0 - mean, d1 = t1 - mean;
    red[tid] = d0 * d0 + d1 * d1;
    __syncthreads();
    for (int s = 128; s > 0; s >>= 1) {
        if (tid < s) red[tid] += red[tid + s];
        __syncthreads();
    }
    const float rs = rsqrtf(red[0] * (1.0f / CH) + 1e-5f);

    Hn[off + tid]       = fmaxf(d0 * rs * gw[tid]       + bw[tid],       0.0f);
    Hn[off + tid + 256] = fmaxf(d1 * rs * gw[tid + 256] + bw[tid + 256], 0.0f);
}

// ---------------------------------------------------------------------------
// Un-permute: out[node_of[row]] = H[row] for valid rows.
// ---------------------------------------------------------------------------
__global__ __launch_bounds__(128)
void gather_out_kernel(const float* __restrict__ H, const int* __restrict__ node_of,
                       float* __restrict__ out)
{
    const int row = blockIdx.x;
    const int nd  = node_of[row];
    if (nd < 0) return;
    ((float4*)out)[(long)nd * (CH / 4) + threadIdx.x] =
        ((const float4*)H)[(long)row * (CH / 4) + threadIdx.x];
}

// ---------------------------------------------------------------------------
extern "C" void kernel_launch(void* const* d_in, const int* in_sizes, int n_in,
                              void* d_out, int out_size, void* d_ws, size_t ws_size,
                              hipStream_t stream)
{
    (void)n_in; (void)out_size; (void)ws_size;
    const float* x     = (const float*)d_in[0];
    const int*   batch = (const int*)  d_in[1];
    const float* fc_w  = (const float*)d_in[2];
    const float* fc_b  = (const float*)d_in[3];
    const float* ln0_g = (const float*)d_in[4];
    const float* ln0_b = (const float*)d_in[5];
    const float* qw    = (const float*)d_in[6];
    const float* kw    = (const float*)d_in[7];
    const float* vw    = (const float*)d_in[8];
    const float* bn_g  = (const float*)d_in[9];
    const float* bn_b  = (const float*)d_in[10];
    const int N = in_sizes[1];

    // ---- workspace carve ----
    char* w = (char*)d_ws;
    const size_t SZ = (size_t)ROWS * CH * sizeof(float);   // 151 MB per dense
    float* H0   = (float*)w; w += SZ;
    float* H1   = (float*)w; w += SZ;
    float* Qb   = (float*)w; w += SZ;
    float* Kb   = (float*)w; w += SZ;   // reused as S after ksum/kvs consume K
    float* Vb   = (float*)w; w += SZ;
    float* KVS  = (float*)w; w += (size_t)G_NUM * CH * CH * sizeof(float);
    float* KSUM = (float*)w; w += (size_t)G_NUM * CH * sizeof(float);
    float* PART = (float*)w; w += 2048 * sizeof(float);
    float* SCAL = (float*)w; w += 256;
    int*   NODE = (int*)w;   w += (size_t)ROWS * sizeof(int);
    int*   CNT  = (int*)w;   w += 256;

    // ---- index build ----
    build_index_kernel<<<G_NUM, 256, 0, stream>>>(batch, N, NODE, CNT);

    const dim3 gFlat(ROWS / 64, CH / 256, 1);

    // ---- fc GEMM (gathered A) + LN + ReLU ----
    gemm_wmma_f32<1><<<gFlat, 256, 0, stream>>>(
        x, fc_w, Qb, ROWS, CH, CIN, CIN, 0, 0, 0, NODE);
    ln_relu_kernel<<<ROWS, 256, 0, stream>>>(Qb, fc_b, ln0_g, ln0_b, H0, 1);

    float* h  = H0;
    float* hn = H1;
    for (int l = 0; l < NLAYER; ++l) {
        const float* qwl = qw + (size_t)l * CH * CH;
        const float* kwl = kw + (size_t)l * CH * CH;
        const float* vwl = vw + (size_t)l * CH * CH;

        // projections
        gemm_wmma_f32<0><<<gFlat, 256, 0, stream>>>(
            h, qwl, Qb, ROWS, CH, CH, CH, 0, 0, 0, nullptr);
        gemm_wmma_f32<0><<<gFlat, 256, 0, stream>>>(
            h, kwl, Kb, ROWS, CH, CH, CH, 0, 0, 0, nullptr);
        gemm_wmma_f32<0><<<gFlat, 256, 0, stream>>>(
            h, vwl, Vb, ROWS, CH, CH, CH, 0, 0, 0, nullptr);

        maskv_kernel<<<ROWS, 128, 0, stream>>>(Vb, NODE);

        // global Frobenius norms of Q and K -> scl = 1/(|Q| * |K|)
        sumsq2_kernel<<<1024, 256, 0, stream>>>(Qb, Kb, PART);
        finalize_scale_kernel<<<1, 256, 0, stream>>>(PART, SCAL);

        // ks_sum (unmasked)
        ksum_kernel<<<dim3(G_NUM, CH / 256), 256, 0, stream>>>(Kb, KSUM);

        // kvs[g] = K[g]^T @ V[g]   (batched, transposed-A)
        gemm_wmma_f32<2><<<dim3(CH / 64, CH / 256, G_NUM), 256, 0, stream>>>(
            Kb, Vb, KVS, CH, CH, MAXN, CH,
            (long)MAXN * CH, (long)MAXN * CH, (long)CH * CH, nullptr);

        // S[g] = Q[g] @ kvs[g]  (S overwrites the K buffer; K is dead now)
        gemm_wmma_f32<0><<<dim3(MAXN / 64, CH / 256, G_NUM), 256, 0, stream>>>(
            Qb, KVS, Kb, MAXN, CH, CH, CH,
            (long)MAXN * CH, (long)CH * CH, (long)MAXN * CH, nullptr);

        // combine + residual + LN + ReLU
        attn_ln_kernel<<<ROWS, 256, 0, stream>>>(
            Kb, Qb, Vb, h, KSUM, SCAL, bn_g + (size_t)l * CH, bn_b + (size_t)l * CH, hn);

        float* t = h; h = hn; hn = t;
    }

    gather_out_kernel<<<ROWS, 128, 0, stream>>>(h, NODE, (float*)d_out);
}